// NeuralMMMModel_56281251447317
// MI455X (gfx1250) — compile-verified
//
#include <hip/hip_runtime.h>
#include <hip/hip_bf16.h>
#include <math.h>

// Problem constants from the reference
#define Bsz 2048
#define Tsz 512
#define Csz 128
#define NCsz 10

typedef __attribute__((ext_vector_type(4))) float f4;
typedef __attribute__((ext_vector_type(2))) float v2f;
typedef __attribute__((ext_vector_type(8))) float v8f;

__device__ __forceinline__ float sigmoidf_(float x) { return 1.0f / (1.0f + expf(-x)); }
__device__ __forceinline__ float gelu_(float x) {
    return 0.5f * x * (1.0f + erff(x * 0.7071067811865475f));
}

// ---------------------------------------------------------------------------
// Kernel 1: adstock weighted reduction over T (memory-bound streaming pass)
//   grid = B blocks, 128 threads (4 waves). Wave j handles T-chunk j (128
//   steps); each lane owns 4 consecutive channels (float4) -> one contiguous
//   512B NT load per wave per timestep. Chunk partials are rescaled by
//   d^(T - chunk_end) and combined in LDS; wave 0 applies saturation and
//   writes last_t[B,C]; wave 1 computes control_feat[B,32].
// ---------------------------------------------------------------------------
__global__ __launch_bounds__(128) void adstock_kernel(
    const float* __restrict__ spend,   // [B,T,C]
    const float* __restrict__ cv,      // [B,NC]
    const float* __restrict__ decay,   // [C]
    const float* __restrict__ alpha,   // [C]
    const float* __restrict__ beta,    // [C]
    const float* __restrict__ Wc,      // [NC,32]
    const float* __restrict__ bc,      // [32]
    float* __restrict__ lastT,         // [B,C]  (ws)
    float* __restrict__ cf)            // [B,32] (ws)
{
    __shared__ float part[4][Csz];

    const int b    = blockIdx.x;
    const int lane = threadIdx.x & 31;
    const int j    = threadIdx.x >> 5;       // chunk id 0..3
    const int c0   = lane * 4;

    f4 dv = *(const f4*)(decay + c0);
    f4 d;
    d.x = sigmoidf_(dv.x); d.y = sigmoidf_(dv.y);
    d.z = sigmoidf_(dv.z); d.w = sigmoidf_(dv.w);

    const float* base = spend + (size_t)b * Tsz * Csz + (size_t)j * 128 * Csz + c0;

    f4 acc = {0.f, 0.f, 0.f, 0.f};
#pragma unroll 8
    for (int t = 0; t < 128; ++t) {
        f4 x = __builtin_nontemporal_load((const f4*)(base + (size_t)t * Csz));
        acc = x + d * acc;
    }

    // weight = d^((3-j)*128); d^128 via 7 squarings
    f4 w = d;
#pragma unroll
    for (int i = 0; i < 7; ++i) w = w * w;   // d^128
    f4 weight;
    if (j == 3)      weight = (f4){1.f, 1.f, 1.f, 1.f};
    else if (j == 2) weight = w;
    else if (j == 1) weight = w * w;
    else             weight = w * w * w;
    acc = acc * weight;

    *(f4*)(&part[j][c0]) = acc;
    __syncthreads();

    if (j == 0) {
        f4 p = *(f4*)(&part[0][c0]) + *(f4*)(&part[1][c0]) +
               *(f4*)(&part[2][c0]) + *(f4*)(&part[3][c0]);
        f4 al = *(const f4*)(alpha + c0);
        f4 be = *(const f4*)(beta + c0);
        f4 r;
        r.x = 2.f * sigmoidf_(al.x) * (1.f - expf(-fmaxf(be.x, 0.01f) * p.x));
        r.y = 2.f * sigmoidf_(al.y) * (1.f - expf(-fmaxf(be.y, 0.01f) * p.y));
        r.z = 2.f * sigmoidf_(al.z) * (1.f - expf(-fmaxf(be.z, 0.01f) * p.z));
        r.w = 2.f * sigmoidf_(al.w) * (1.f - expf(-fmaxf(be.w, 0.01f) * p.w));
        *(f4*)(lastT + (size_t)b * Csz + c0) = r;
    } else if (j == 1) {
        // control_feat[b, lane] = cv[b,:] @ Wc[:,lane] + bc[lane]
        float s = bc[lane];
#pragma unroll
        for (int i = 0; i < NCsz; ++i) s += cv[b * NCsz + i] * Wc[i * 32 + lane];
        cf[b * 32 + lane] = s;
    }
}

// ---------------------------------------------------------------------------
// WMMA fragment helpers for V_WMMA_F32_16X16X4_F32 (wave32).
//  A 16x4 f32: lanes 0-15 -> M=lane, VGPR0=K0,VGPR1=K1; lanes 16-31 -> K2/K3.
//  B 4x16 f32: same pattern with N instead of M.
//  C/D 16x16 f32: v8f; VGPR r -> M=r (lanes 0-15), M=r+8 (lanes 16-31), N=lane&15.
// ---------------------------------------------------------------------------

// LDS row strides padded (+1) to spread banks.
#define A_LD   129
#define H_LD   257
#define CB_LD  161
#define O_LD   65

__global__ __launch_bounds__(256) void mlp_kernel(
    const float* __restrict__ lastT,  // [B,128] (ws)
    const float* __restrict__ cf,     // [B,32]  (ws)
    const float* __restrict__ W1,     // [128,256]
    const float* __restrict__ b1,     // [256]
    const float* __restrict__ W2,     // [256,128]
    const float* __restrict__ b2,     // [128]
    const float* __restrict__ Wo1,    // [160,64]
    const float* __restrict__ bo1,    // [64]
    const float* __restrict__ Wo2,    // [64,1]
    const float* __restrict__ bo2,    // [1]
    float* __restrict__ out)          // [B]
{
    __shared__ float a_lds[16 * A_LD];     // last_t tile  16x128
    __shared__ float h_lds[16 * H_LD];     // hidden       16x256
    __shared__ float c_lds[16 * CB_LD];    // combined     16x160
    __shared__ float o_lds[16 * O_LD];     // out hidden   16x64

    const int tid  = threadIdx.x;
    const int lane = tid & 31;
    const int w    = tid >> 5;             // wave 0..7
    const int m0   = blockIdx.x * 16;

    const int mrow = lane & 15;            // local M row of the A fragment
    const int ksel = (lane >> 4) << 1;     // 0 for lanes 0-15, 2 for 16-31
    const int ncol = lane & 15;            // local N col of B / C

    // Stage A tile (16x128) into LDS
    for (int i = tid; i < 16 * 128; i += 256) {
        int m = i >> 7, k = i & 127;
        a_lds[m * A_LD + k] = lastT[(size_t)(m0 + m) * Csz + k];
    }
    __syncthreads();

    // ---- GEMM1: h = gelu(A[16x128] @ W1[128x256] + b1) ; 2 N-tiles per wave
    {
        v8f acc0 = {}; v8f acc1 = {};
        const int n0a = w * 16, n0b = (w + 8) * 16;
#pragma unroll 4
        for (int kb = 0; kb < 128; kb += 4) {
            int kk = kb + ksel;
            v2f a;  a.x  = a_lds[mrow * A_LD + kk];      a.y  = a_lds[mrow * A_LD + kk + 1];
            v2f ba; ba.x = W1[kk * 256 + n0a + ncol];    ba.y = W1[(kk + 1) * 256 + n0a + ncol];
            v2f bb; bb.x = W1[kk * 256 + n0b + ncol];    bb.y = W1[(kk + 1) * 256 + n0b + ncol];
            acc0 = __builtin_amdgcn_wmma_f32_16x16x4_f32(false, a, false, ba, (short)0, acc0, false, false);
            acc1 = __builtin_amdgcn_wmma_f32_16x16x4_f32(false, a, false, bb, (short)0, acc1, false, false);
        }
#pragma unroll
        for (int r = 0; r < 8; ++r) {
            int m = (lane < 16) ? r : (r + 8);
            h_lds[m * H_LD + n0a + ncol] = gelu_(acc0[r] + b1[n0a + ncol]);
            h_lds[m * H_LD + n0b + ncol] = gelu_(acc1[r] + b1[n0b + ncol]);
        }
    }
    __syncthreads();

    // ---- GEMM2: interactions = h[16x256] @ W2[256x128] + b2 ; 1 N-tile per wave
    {
        v8f acc = {};
        const int n0 = w * 16;
#pragma unroll 4
        for (int kb = 0; kb < 256; kb += 4) {
            int kk = kb + ksel;
            v2f a;  a.x  = h_lds[mrow * H_LD + kk];   a.y  = h_lds[mrow * H_LD + kk + 1];
            v2f bb; bb.x = W2[kk * 128 + n0 + ncol];  bb.y = W2[(kk + 1) * 128 + n0 + ncol];
            acc = __builtin_amdgcn_wmma_f32_16x16x4_f32(false, a, false, bb, (short)0, acc, false, false);
        }
#pragma unroll
        for (int r = 0; r < 8; ++r) {
            int m = (lane < 16) ? r : (r + 8);
            c_lds[m * CB_LD + n0 + ncol] = acc[r] + b2[n0 + ncol];
        }
    }
    // concat control features into combined[:,128:160]
    for (int i = tid; i < 16 * 32; i += 256) {
        int m = i >> 5, k = i & 31;
        c_lds[m * CB_LD + 128 + k] = cf[(size_t)(m0 + m) * 32 + k];
    }
    __syncthreads();

    // ---- GEMM3: o = gelu(combined[16x160] @ Wo1[160x64] + bo1) ; waves 0..3
    if (w < 4) {
        v8f acc = {};
        const int n0 = w * 16;
#pragma unroll 4
        for (int kb = 0; kb < 160; kb += 4) {
            int kk = kb + ksel;
            v2f a;  a.x  = c_lds[mrow * CB_LD + kk];  a.y  = c_lds[mrow * CB_LD + kk + 1];
            v2f bb; bb.x = Wo1[kk * 64 + n0 + ncol];  bb.y = Wo1[(kk + 1) * 64 + n0 + ncol];
            acc = __builtin_amdgcn_wmma_f32_16x16x4_f32(false, a, false, bb, (short)0, acc, false, false);
        }
#pragma unroll
        for (int r = 0; r < 8; ++r) {
            int m = (lane < 16) ? r : (r + 8);
            o_lds[m * O_LD + n0 + ncol] = gelu_(acc[r] + bo1[n0 + ncol]);
        }
    }
    __syncthreads();

    // ---- Final: out = o[16x64] @ Wo2[64] + bo2
    if (tid < 16) {
        float s = 0.f;
#pragma unroll 8
        for (int j = 0; j < 64; ++j) s += o_lds[tid * O_LD + j] * Wo2[j];
        out[m0 + tid] = s + bo2[0];
    }
}

// ---------------------------------------------------------------------------
extern "C" void kernel_launch(void* const* d_in, const int* in_sizes, int n_in,
                              void* d_out, int out_size, void* d_ws, size_t ws_size,
                              hipStream_t stream) {
    const float* spend = (const float*)d_in[0];   // [B,T,C]
    const float* cv    = (const float*)d_in[1];   // [B,NC]
    const float* decay = (const float*)d_in[2];   // [C]
    const float* alpha = (const float*)d_in[3];   // [C]
    const float* beta  = (const float*)d_in[4];   // [C]
    const float* W1    = (const float*)d_in[5];   // [C,2C]
    const float* b1    = (const float*)d_in[6];   // [2C]
    const float* W2    = (const float*)d_in[7];   // [2C,C]
    const float* b2    = (const float*)d_in[8];   // [C]
    const float* Wc    = (const float*)d_in[9];   // [NC,32]
    const float* bc    = (const float*)d_in[10];  // [32]
    const float* Wo1   = (const float*)d_in[11];  // [C+32,64]
    const float* bo1   = (const float*)d_in[12];  // [64]
    const float* Wo2   = (const float*)d_in[13];  // [64,1]
    const float* bo2   = (const float*)d_in[14];  // [1]

    float* lastT = (float*)d_ws;                        // B*C floats
    float* cf    = (float*)d_ws + (size_t)Bsz * Csz;    // B*32 floats

    adstock_kernel<<<Bsz, 128, 0, stream>>>(spend, cv, decay, alpha, beta, Wc, bc,
                                            lastT, cf);
    mlp_kernel<<<Bsz / 16, 256, 0, stream>>>(lastT, cf, W1, b1, W2, b2,
                                             Wo1, bo1, Wo2, bo2, (float*)d_out);
}